// OneHotVectorEncoding_62843961475696
// MI455X (gfx1250) — compile-verified
//
#include <hip/hip_runtime.h>

// One-hot encoding: out[b, s, x[b,s]] = 1.0f, zeros elsewhere.
// Shape: rows = B*S = 4096, V = 32000. Output 524 MB fp32.
//
// Strategy (MI455X / gfx1250):
//  - Store-bandwidth bound: floor = 524 MB / 23.3 TB/s ~= 22.5 us.
//  - Single fused pass: each block owns one contiguous 128 KB row.
//  - float4 (b128) non-temporal stores: output (524 MB) >> L2 (192 MB),
//    so TH=NT avoids thrashing the global L2 for write-once data.
//  - wave32: 256-thread block = 8 waves; 4096 blocks saturate all WGPs.

typedef __attribute__((ext_vector_type(4))) float v4f;

__global__ __launch_bounds__(256) void onehot_row_kernel(
    const int* __restrict__ x,   // [rows] indices
    float* __restrict__ out,     // [rows * V]
    int V) {
  const int row = blockIdx.x;

  // Uniform (block-scalar) index load; prefetch the tiny index array
  // (lowers to global_prefetch_b8 on gfx1250).
  __builtin_prefetch(&x[row], 0, 3);
  const int idx = x[row];                 // uniform -> scalar load path

  const int V4      = V >> 2;             // float4s per row (8000)
  const int target4 = idx >> 2;           // which float4 holds the 1.0
  const int comp    = idx & 3;            // which component within it

  v4f* __restrict__ out4 = (v4f*)(out + (size_t)row * (size_t)V);

  const v4f zero = (v4f)0.0f;

  // Stream the row with 16-byte non-temporal stores.
  for (int p = threadIdx.x; p < V4; p += blockDim.x) {
    v4f v = zero;
    if (p == target4) {                   // rare: once per row
      v.x = (comp == 0) ? 1.0f : 0.0f;
      v.y = (comp == 1) ? 1.0f : 0.0f;
      v.z = (comp == 2) ? 1.0f : 0.0f;
      v.w = (comp == 3) ? 1.0f : 0.0f;
    }
    __builtin_nontemporal_store(v, out4 + p);
  }

  // Tail for V not divisible by 4 (not hit for V=32000; kept for generality).
  for (int p = (V4 << 2) + (int)threadIdx.x; p < V; p += blockDim.x) {
    float f = (p == idx) ? 1.0f : 0.0f;
    __builtin_nontemporal_store(f, out + (size_t)row * (size_t)V + p);
  }
}

extern "C" void kernel_launch(void* const* d_in, const int* in_sizes, int n_in,
                              void* d_out, int out_size, void* d_ws, size_t ws_size,
                              hipStream_t stream) {
  (void)n_in; (void)d_ws; (void)ws_size;

  const int* x = (const int*)d_in[0];   // [B*S] indices
  float* out   = (float*)d_out;         // [B*S*V] fp32

  const int rows = in_sizes[0];         // B*S = 4096
  const int V    = out_size / rows;     // 32000

  dim3 grid((unsigned)rows);
  dim3 block(256);
  onehot_row_kernel<<<grid, block, 0, stream>>>(x, out, V);
}